// Attention_4939212391217
// MI455X (gfx1250) — compile-verified
//
#include <hip/hip_runtime.h>
#include <cstdint>

typedef __attribute__((ext_vector_type(2))) float v2f;
typedef __attribute__((ext_vector_type(8))) float v8f;

#define BATCH 2
#define SEQ   2048
#define NXDIM 1024
#define NHEAD 16
#define DK    64

#define BM 128
#define BN 128
#define BK 16

static __device__ __forceinline__ v8f wmma_f32(v2f a, v2f b, v8f c) {
    // D = A(16x4 f32) * B(4x16 f32) + C(16x16 f32)
    return __builtin_amdgcn_wmma_f32_16x16x4_f32(
        /*neg_a=*/false, a, /*neg_b=*/false, b,
        /*c_mod=*/(short)0, c, /*reuse_a=*/false, /*reuse_b=*/false);
}

// Async global->LDS copy of 16B per lane (CDNA5 GLOBAL_LOAD_ASYNC_TO_LDS_B128,
// tracked by ASYNCcnt). ldsoff is the wave-relative LDS byte address (low 32
// bits of the generic pointer to a __shared__ object). The instruction offset
// is added to BOTH the memory and LDS addresses.
static __device__ __forceinline__ void async_b128(unsigned ldsoff, const void* gptr) {
    asm volatile("global_load_async_to_lds_b128 %0, %1, off"
                 :: "v"(ldsoff), "v"((unsigned long long)(uintptr_t)gptr)
                 : "memory");
}
static __device__ __forceinline__ void async_b128_o16(unsigned ldsoff, const void* gptr) {
    asm volatile("global_load_async_to_lds_b128 %0, %1, off offset:16"
                 :: "v"(ldsoff), "v"((unsigned long long)(uintptr_t)gptr)
                 : "memory");
}
static __device__ __forceinline__ void wait_async0() {
    asm volatile("s_wait_asynccnt 0x0" ::: "memory");
}
static __device__ __forceinline__ unsigned lds_off(const void* p) {
    return (unsigned)(uintptr_t)p;
}

// ---------------------------------------------------------------------------
// Fused GEMM: C[M,N] = A[M,K] @ W[K,N] + bias[N], fp32, row-major.
// Double-buffered LDS: A tile staged via async global->LDS B128 copies,
// B tile staged via registers (k-pair interleave transform: Bs2[k/2][2n+(k&1)]
// so each WMMA B-fragment (B[k][n], B[k+1][n]) is one aligned ds_load_b64).
// Output columns routed into up to 3 destinations in chunks of `nsplit`.
// ---------------------------------------------------------------------------
__global__ __launch_bounds__(256, 2)
void gemm_bias_split(const float* __restrict__ A, const float* __restrict__ W,
                     const float* __restrict__ bias,
                     float* __restrict__ d0, float* __restrict__ d1,
                     float* __restrict__ d2,
                     int M, int N, int K, int nsplit, int ldc)
{
    __shared__ float As[2][BM][BK + 4];           // pitch 20: b64 frags, conflict-free
    __shared__ float Bs2[2][BK / 2][2 * BN + 8];  // pitch 264: k-pair interleaved

    const int tid  = threadIdx.x;
    const int lane = tid & 31;
    const int lo   = lane & 15;
    const int hi   = lane >> 4;
    const int m0   = (tid >> 5) * 16;
    const int tM   = blockIdx.y * BM;
    const int tN   = blockIdx.x * BN;

    const int lr = tid >> 1;                      // A tile: 128 rows x 16 cols, 8 floats/thread
    const int lc = (tid & 1) * 8;
    const int kp = tid >> 5;                      // B tile: k-pair 0..7
    const int nb = (tid & 31) * 4;                // 4 columns per thread

    const float* arow = A + (size_t)(tM + lr) * K + lc;  // advance by k0
    v8f acc[8] = {};                              // 16 x 128 fp32 accumulators per wave

    // ---- prologue: stage tile 0 into buffer 0 ----
    async_b128(lds_off(&As[0][lr][lc]), arow);
    async_b128_o16(lds_off(&As[0][lr][lc]), arow);
    {
        const float* w0 = W + (size_t)(2 * kp) * N + tN + nb;
        float4 r0 = *(const float4*)w0;
        float4 r1 = *(const float4*)(w0 + N);
        float4 s0; s0.x = r0.x; s0.y = r1.x; s0.z = r0.y; s0.w = r1.y;
        float4 s1; s1.x = r0.z; s1.y = r1.z; s1.z = r0.w; s1.w = r1.w;
        *(float4*)&Bs2[0][kp][2 * nb]     = s0;
        *(float4*)&Bs2[0][kp][2 * nb + 4] = s1;
    }
    wait_async0();
    __syncthreads();

    for (int k0 = 0; k0 < K; k0 += BK) {
        const int  buf  = (k0 >> 4) & 1;
        const bool more = (k0 + BK) < K;

        // ---- prefetch tile k0+BK into the other buffer ----
        float4 r0 = {}, r1 = {};
        if (more) {
            const float* an = arow + k0 + BK;
            async_b128(lds_off(&As[buf ^ 1][lr][lc]), an);
            async_b128_o16(lds_off(&As[buf ^ 1][lr][lc]), an);
            const float* w0 = W + (size_t)(k0 + BK + 2 * kp) * N + tN + nb;
            r0 = *(const float4*)w0;
            r1 = *(const float4*)(w0 + N);
        }

        // ---- compute on current buffer ----
        #pragma unroll
        for (int kk = 0; kk < BK; kk += 4) {
            const int cp = (kk >> 1) + hi;                 // interleaved pair-row
            v2f a = *(const v2f*)&As[buf][m0 + lo][kk + hi * 2];
            #pragma unroll
            for (int j = 0; j < 8; ++j) {
                v2f b = *(const v2f*)&Bs2[buf][cp][2 * (j * 16 + lo)];
                acc[j] = wmma_f32(a, b, acc[j]);
            }
        }

        if (more) {
            float4 s0; s0.x = r0.x; s0.y = r1.x; s0.z = r0.y; s0.w = r1.y;
            float4 s1; s1.x = r0.z; s1.y = r1.z; s1.z = r0.w; s1.w = r1.w;
            *(float4*)&Bs2[buf ^ 1][kp][2 * nb]     = s0;
            *(float4*)&Bs2[buf ^ 1][kp][2 * nb + 4] = s1;
            wait_async0();
        }
        __syncthreads();
    }

    const int rowbase = tM + m0 + hi * 8;         // C layout: VGPR r -> row r + 8*hi, col = lo
    #pragma unroll
    for (int j = 0; j < 8; ++j) {
        const int col = tN + j * 16 + lo;
        const float bv = bias[col];
        const int seg = (col >= nsplit) ? ((col >= 2 * nsplit) ? 2 : 1) : 0;
        float* dst = (seg == 0) ? d0 : ((seg == 1) ? d1 : d2);
        const int c2 = col - seg * nsplit;
        #pragma unroll
        for (int r = 0; r < 8; ++r) {
            dst[(size_t)(rowbase + r) * ldc + c2] = acc[j][r] + bv;
        }
    }
}

// ---------------------------------------------------------------------------
// Causal multi-head attention, flash-style online softmax.
// q,k,v,o: [B, S, NX] fp32 (head h lives in columns [h*64, h*64+64)).
// Block = 256 threads = 8 waves; block handles (b, h, 128 q-rows);
// each wave owns a 16-row q tile. Matches reference: mask with -1e9 FIRST,
// then scale by 1/sqrt(64)=0.125, then softmax.
// K tile staged with async global->LDS; V staged k-pair interleaved.
// ---------------------------------------------------------------------------
__global__ __launch_bounds__(256, 2)
void attn_causal(const float* __restrict__ q, const float* __restrict__ kmat,
                 const float* __restrict__ vmat, float* __restrict__ o)
{
    __shared__ float Ks[16][DK + 4];           // pitch 68 (row-major: frags already adjacent)
    __shared__ float Vs2[8][2 * DK + 8];       // pitch 136: k-pair interleaved
    __shared__ float Ps[8][16][18];            // pitch 18: keeps P A-frag reads b64-aligned

    int id = blockIdx.x;
    const int qblk = id % (SEQ / 128); id /= (SEQ / 128);
    const int h = id % NHEAD;
    const int b = id / NHEAD;
    const int q0 = qblk * 128;

    const int tid  = threadIdx.x;
    const int wave = tid >> 5;
    const int lane = tid & 31;
    const int lo   = lane & 15;
    const int hi   = lane >> 4;
    const int qrow0 = q0 + wave * 16;          // wave's global q rows [qrow0, qrow0+16)

    const float* qbase = q    + ((size_t)b * SEQ) * NXDIM + h * DK;
    const float* kbase = kmat + ((size_t)b * SEQ) * NXDIM + h * DK;
    const float* vbase = vmat + ((size_t)b * SEQ) * NXDIM + h * DK;

    // Preload Q A-fragments for all 16 k-steps (K-dim = 64 = 16 * 4).
    v2f afrag[16];
    #pragma unroll
    for (int s = 0; s < 16; ++s) {
        const float2 qv = *(const float2*)(qbase + (size_t)(qrow0 + lo) * NXDIM + s * 4 + hi * 2);
        afrag[s].x = qv.x;
        afrag[s].y = qv.y;
    }

    v8f oacc[4] = {};                          // 16 x 64 output accumulators
    float Mrow[8], Lrow[8];
    #pragma unroll
    for (int r = 0; r < 8; ++r) { Mrow[r] = -INFINITY; Lrow[r] = 0.0f; }

    const int ktmax_wave = (qrow0 + 15) >> 4;  // inclusive diagonal tile for this wave
    const int ktmax_blk  = (q0 + 127) >> 4;

    const int klr = tid >> 4;                  // K tile: 16 rows x 64 cols, 16B/thread (async)
    const int klc = (tid & 15) * 4;
    const int vp  = tid >> 5;                  // V tile: row-pair 0..7
    const int vc  = (tid & 31) * 2;            // 2 columns per thread

    const unsigned ks_dst = lds_off(&Ks[klr][klc]);

    for (int kt = 0; kt <= ktmax_blk; ++kt) {
        // Cooperative K/V tile load (rows kt*16 .. kt*16+15)
        {
            async_b128(ks_dst, kbase + (size_t)(kt * 16 + klr) * NXDIM + klc);
            const float* vsrc = vbase + (size_t)(kt * 16 + 2 * vp) * NXDIM + vc;
            float2 v0 = *(const float2*)vsrc;
            float2 v1 = *(const float2*)(vsrc + NXDIM);
            float4 sv; sv.x = v0.x; sv.y = v1.x; sv.z = v0.y; sv.w = v1.y;
            *(float4*)&Vs2[vp][2 * vc] = sv;
            wait_async0();
        }
        __syncthreads();

        if (kt <= ktmax_wave) {                // wave-uniform: EXEC stays all-ones
            // S = Q @ K^T  (16x16 scores)
            v8f s = {};
            #pragma unroll
            for (int st = 0; st < 16; ++st) {
                v2f bf = *(const v2f*)&Ks[lo][st * 4 + hi * 2];   // B[k][n] = K[n][k]
                s = wmma_f32(afrag[st], bf, s);
            }

            // Causal mask (-1e9) then scale by 0.125 (matches reference order)
            const int kc = kt * 16 + lo;
            #pragma unroll
            for (int r = 0; r < 8; ++r) {
                const int qr = qrow0 + r + hi * 8;
                float v = s[r];
                v = (kc <= qr) ? v : -1.0e9f;
                s[r] = v * 0.125f;
            }

            // Row max over the 16 lanes of this half-wave
            float alpha[8];
            #pragma unroll
            for (int r = 0; r < 8; ++r) {
                float v = s[r];
                #pragma unroll
                for (int off = 8; off >= 1; off >>= 1)
                    v = fmaxf(v, __shfl_xor(v, off, 32));
                const float mnew = fmaxf(Mrow[r], v);
                alpha[r] = __expf(Mrow[r] - mnew);
                Mrow[r] = mnew;
            }

            // P = exp(s - M); row sums; rescale running state
            #pragma unroll
            for (int r = 0; r < 8; ++r) {
                s[r] = __expf(s[r] - Mrow[r]);
                float v = s[r];
                #pragma unroll
                for (int off = 8; off >= 1; off >>= 1)
                    v += __shfl_xor(v, off, 32);
                Lrow[r] = Lrow[r] * alpha[r] + v;
            }
            #pragma unroll
            for (int j = 0; j < 4; ++j)
                #pragma unroll
                for (int r = 0; r < 8; ++r)
                    oacc[j][r] *= alpha[r];

            // Stage P in LDS to redistribute C-layout -> A-fragment layout
            #pragma unroll
            for (int r = 0; r < 8; ++r)
                Ps[wave][r + hi * 8][lo] = s[r];
            asm volatile("" ::: "memory");     // same-wave LDS is in-order; keep compiler ordered

            // O += P @ V   (k-dim = 16 -> 4 wmma steps x 4 column tiles)
            #pragma unroll
            for (int kb = 0; kb < 16; kb += 4) {
                const int cp = (kb >> 1) + hi;
                v2f pa = *(const v2f*)&Ps[wave][lo][kb + hi * 2];
                #pragma unroll
                for (int j = 0; j < 4; ++j) {
                    v2f bv = *(const v2f*)&Vs2[cp][2 * (j * 16 + lo)];
                    oacc[j] = wmma_f32(pa, bv, oacc[j]);
                }
            }
        }
        __syncthreads();
    }

    // Normalize and write out
    float* obase = o + ((size_t)b * SEQ) * NXDIM + h * DK;
    #pragma unroll
    for (int r = 0; r < 8; ++r) {
        const float inv = 1.0f / Lrow[r];
        const int row = qrow0 + r + hi * 8;
        #pragma unroll
        for (int j = 0; j < 4; ++j)
            obase[(size_t)row * NXDIM + j * 16 + lo] = oacc[j][r] * inv;
    }
}

// ---------------------------------------------------------------------------
extern "C" void kernel_launch(void* const* d_in, const int* in_sizes, int n_in,
                              void* d_out, int out_size, void* d_ws, size_t ws_size,
                              hipStream_t stream) {
    (void)in_sizes; (void)n_in; (void)out_size; (void)ws_size;

    const float* x      = (const float*)d_in[0];   // [B,S,NX]
    const float* w_attn = (const float*)d_in[1];   // [NX, 3NX]
    const float* b_attn = (const float*)d_in[2];   // [3NX]
    const float* w_proj = (const float*)d_in[3];   // [NX, NX]
    const float* b_proj = (const float*)d_in[4];   // [NX]

    float* out = (float*)d_out;
    const size_t ASZ = (size_t)BATCH * SEQ * NXDIM;    // 4,194,304
    float* k_out = out + ASZ;                          // present[0] = k
    float* v_out = out + ASZ + ASZ;                    // present[1] = v

    float* q_ws = (float*)d_ws;                        // [B,S,NX] 16 MB
    float* o_ws = q_ws + ASZ;                          // [B,S,NX] 16 MB

    const int M = BATCH * SEQ;                         // 4096
    dim3 blk(256);

    // 1) qkv = x @ w_attn + b_attn, split into q(ws), k,v(d_out present)
    gemm_bias_split<<<dim3(3 * NXDIM / BN, M / BM), blk, 0, stream>>>(
        x, w_attn, b_attn, q_ws, k_out, v_out,
        M, 3 * NXDIM, NXDIM, /*nsplit=*/NXDIM, /*ldc=*/NXDIM);

    // 2) causal attention -> o_ws
    attn_causal<<<dim3(BATCH * NHEAD * (SEQ / 128)), blk, 0, stream>>>(
        q_ws, k_out, v_out, o_ws);

    // 3) a = o @ w_proj + b_proj -> d_out[0:ASZ]
    gemm_bias_split<<<dim3(NXDIM / BN, M / BM), blk, 0, stream>>>(
        o_ws, w_proj, b_proj, out, out, out,
        M, NXDIM, NXDIM, /*nsplit=*/1 << 28, /*ldc=*/NXDIM);
}